// MultiRegionalFusionModule_51049981281334
// MI455X (gfx1250) — compile-verified
//
#include <hip/hip_runtime.h>
#include <math.h>

// MultiRegionalFusion on MI455X (gfx1250, wave32, WMMA).
//
// Roofline: 2.15 GFLOP vs 256 MB HBM traffic -> ~11us floor @ 23.3 TB/s.
// FP32 VALU would need ~1G lane-FMAs (compute bound), so the contraction runs
// on V_WMMA_F32_16X16X4_F32: M = 16 s (A loads coalesced from regional),
// N = 16 t (B = attn, register-resident), k accumulated in 16 steps of 4.
//
// v2 changes (latency reasoning, compile-only):
//  * A-operands for a full k-sweep (32 loads) are batched into registers
//    before the 32 WMMAs -> up to 32 outstanding loads/wave instead of 2
//    (previous asm showed `s_wait_loadcnt 0x0` after every 2 loads).
//  * 4 waves share one 128-thread workgroup and one 34KB LDS transpose tile
//    (each wave owns 4 of the 16 f values) -> 8192 waves total and better
//    waves/SIMD, since LDS per WG is amortized 4x.

typedef __attribute__((ext_vector_type(2))) float v2f;
typedef __attribute__((ext_vector_type(8))) float v8f;

#define BB 8
#define TT 32
#define SS 512
#define FF 128
#define KK 64
#define NN 200

// ---------------------------------------------------------------------------
// Kernel 1: corr double-gather + softmax over K=64 -> attn[b][t][k] in d_ws.
// 256 rows; one thread per (b,t). Tiny (64KB), L2-resident.
// ---------------------------------------------------------------------------
__global__ __launch_bounds__(256) void attn_softmax_kernel(
    const float* __restrict__ corr, const int* __restrict__ tidx,
    const int* __restrict__ ridx, float* __restrict__ attn) {
  const int p = threadIdx.x;               // p = b*T + t, 0..255
  const int row = tidx[p];
  const float* cr = corr + (long)row * NN;

  float m = -3.402823466e38f;
  for (int k = 0; k < KK; ++k) m = fmaxf(m, cr[ridx[k]]);
  float s = 0.0f;
  for (int k = 0; k < KK; ++k) s += __expf(cr[ridx[k]] - m);
  const float inv = 1.0f / s;
  float* o = attn + p * KK;
  for (int k = 0; k < KK; ++k) o[k] = __expf(cr[ridx[k]] - m) * inv;
}

// ---------------------------------------------------------------------------
// Kernel 2: ctx = attn x regional via WMMA f32 16x16x4, + target, transposed
// store. 128 threads = 4 waves per WG; block = (s-tile 16, f-chunk 16, b);
// wave w owns f = fc0 + w*4 .. +3. Shared LDS tile transposes (s,t)->(t,s,f).
// ---------------------------------------------------------------------------
__global__ __launch_bounds__(128) void fuse_wmma_kernel(
    const float* __restrict__ tgt, const float* __restrict__ reg,
    const float* __restrict__ attn, float* __restrict__ out) {
  // 16 s x 32 t x 16 f, f-row padded to 17 words to spread LDS banks
  __shared__ float lds[16 * 32 * 17];

  const int tid   = threadIdx.x;
  const int lane  = tid & 31;
  const int wv    = tid >> 5;    // wave id 0..3 -> owns 4 f values
  const int tlo   = lane & 15;   // A: M index (s); B/D: N index (t)
  const int khalf = lane >> 4;   // A/B: K half select; D: row+8 select
  const int s0  = blockIdx.x * 16;
  const int fc0 = blockIdx.y * 16;
  const int b   = blockIdx.z;

  // --- Preload B fragments (attn) for both t-tiles, all 16 k-steps.
  // B 4x16 f32 layout (mirror of A 16x4): VGPR0 lanes0-15 K=k0, lanes16-31
  // K=k0+2; VGPR1 = +1 k. 64 VGPRs, reused for all 4 f of this wave.
  float bf[2][16][2];
  const float* ab = attn + b * (TT * KK);
#pragma unroll
  for (int tile = 0; tile < 2; ++tile) {
    const int t = tlo + 16 * tile;
#pragma unroll
    for (int ks = 0; ks < 16; ++ks) {
      bf[tile][ks][0] = ab[t * KK + ks * 4 + khalf * 2 + 0];
      bf[tile][ks][1] = ab[t * KK + ks * 4 + khalf * 2 + 1];
    }
  }

  // --- A base pointer: reg[b, k, f, s]; lane supplies s = s0+tlo and the
  // k half (khalf*2). Each load instruction: 2 coalesced 64B segments.
  const float* pb = reg + (size_t)b * (KK * FF * SS)
                        + (size_t)khalf * 2 * (FF * SS)
                        + (size_t)(s0 + tlo);

#pragma unroll
  for (int fi = 0; fi < 4; ++fi) {
    const int fl = wv * 4 + fi;                   // f index within chunk
    const float* pf = pb + (size_t)(fc0 + fl) * SS;

    // Batch the whole k-sweep of A loads first: 32 outstanding loads/wave,
    // consumed below with staged s_wait_loadcnt counts.
    float ax[16], ay[16];
#pragma unroll
    for (int ks = 0; ks < 16; ++ks) {
      ax[ks] = pf[(size_t)ks * (4 * FF * SS)];              // k = 4ks+2*khalf
      ay[ks] = pf[(size_t)ks * (4 * FF * SS) + (FF * SS)];  // k = +1
    }

    v8f d0 = {};   // t-tile 0 (t=0..15)
    v8f d1 = {};   // t-tile 1 (t=16..31)
#pragma unroll
    for (int ks = 0; ks < 16; ++ks) {
      v2f a;  a.x  = ax[ks];         a.y  = ay[ks];
      v2f b0; b0.x = bf[0][ks][0];   b0.y = bf[0][ks][1];
      v2f b1; b1.x = bf[1][ks][0];   b1.y = bf[1][ks][1];
      d0 = __builtin_amdgcn_wmma_f32_16x16x4_f32(
          false, a, false, b0, (short)0, d0, false, false);
      d1 = __builtin_amdgcn_wmma_f32_16x16x4_f32(
          false, a, false, b1, (short)0, d1, false, false);
    }

    // Stash D (s rows x t cols) into the shared transpose tile.
    // VGPR j, lanes0-15: (s=j, t=tlo); lanes16-31: (s=j+8, t=tlo).
#pragma unroll
    for (int j = 0; j < 8; ++j) {
      const int sl = j + 8 * khalf;
      lds[(sl * 32 + tlo) * 17 + fl]      = d0[j];
      lds[(sl * 32 + tlo + 16) * 17 + fl] = d1[j];
    }
  }

  __syncthreads();  // all 4 waves' stashes visible before writeback

  // --- Writeback: out[b,t,s0+sl,fc0+f0..f0+3] = lds + target. 4-lane groups
  // cover 64B-contiguous f segments; target add fused (coalesced float4).
#pragma unroll 4
  for (int it = 0; it < 16; ++it) {
    const int q  = it * 128 + tid;       // 0..2047 float4 slots
    const int f0 = (q & 3) * 4;
    const int r  = q >> 2;               // (sl,t) pair, 0..511
    const int t  = r & 31;
    const int sl = r >> 5;
    const int lb = (sl * 32 + t) * 17 + f0;
    const size_t oi =
        (((size_t)(b * TT + t)) * SS + (size_t)(s0 + sl)) * FF + (fc0 + f0);
    const float4 tv = *(const float4*)(tgt + oi);
    float4 v;
    v.x = lds[lb + 0] + tv.x;
    v.y = lds[lb + 1] + tv.y;
    v.z = lds[lb + 2] + tv.z;
    v.w = lds[lb + 3] + tv.w;
    *(float4*)(out + oi) = v;
  }
}

// ---------------------------------------------------------------------------
extern "C" void kernel_launch(void* const* d_in, const int* in_sizes, int n_in,
                              void* d_out, int out_size, void* d_ws,
                              size_t ws_size, hipStream_t stream) {
  (void)in_sizes; (void)n_in; (void)out_size; (void)ws_size;
  const float* tgt  = (const float*)d_in[0];  // [B,T,S,F]
  const float* reg  = (const float*)d_in[1];  // [B,K,F,S]
  const float* corr = (const float*)d_in[2];  // [N,N]
  const int*   tidx = (const int*)d_in[3];    // [B,T]
  const int*   ridx = (const int*)d_in[4];    // [K]
  float* outp = (float*)d_out;
  float* attn = (float*)d_ws;                 // [B,T,K] = 64KB scratch

  attn_softmax_kernel<<<1, 256, 0, stream>>>(corr, tidx, ridx, attn);

  dim3 grid(SS / 16, FF / 16, BB);            // 2048 blocks x 4 waves
  fuse_wmma_kernel<<<grid, 128, 0, stream>>>(tgt, reg, attn, outp);
}